// DetectionLoss_3839700762848
// MI455X (gfx1250) — compile-verified
//
#include <hip/hip_runtime.h>
#include <hip/hip_bf16.h>

// Problem constants (match reference)
#define BB 16
#define AA 65536
#define GG 32
#define EPSF 1e-7f

typedef __attribute__((ext_vector_type(2))) float v2f;
typedef __attribute__((ext_vector_type(8))) float v8f;

__device__ __forceinline__ float clip0(float x) { return fmaxf(x, 0.f); }

// ---------------------------------------------------------------------------
// K0: zero the per-(image,gt) argmax slots (atomicMax accumulators)
// ---------------------------------------------------------------------------
__global__ void k_init(unsigned long long* __restrict__ best_anchor) {
  int i = threadIdx.x;
  if (i < BB * GG) best_anchor[i] = 0ull;
}

// ---------------------------------------------------------------------------
// K1: per-anchor IoU vs 32 GTs (GTs staged in LDS). Writes best_iou/best_idx
// per anchor, and per-GT best anchor via packed u64 atomicMax:
//   key = (iou_bits << 32) | (0xFFFFFFFF - anchor)   (ties -> lowest anchor)
// iou >= 0 so float bits are order-preserving.
// ---------------------------------------------------------------------------
__global__ void k_match(const float* __restrict__ anchors,
                        const float* __restrict__ gt,
                        float* __restrict__ best_iou,
                        int* __restrict__ best_idx,
                        unsigned long long* __restrict__ best_anchor) {
  const int b = blockIdx.y;
  const int a = blockIdx.x * 256 + threadIdx.x;
  const int tid = threadIdx.x;
  __shared__ float sgt[GG * 4];
  __shared__ unsigned long long slot[GG];
  if (tid < GG * 4) sgt[tid] = gt[(size_t)b * GG * 4 + tid];
  if (tid < GG) slot[tid] = 0ull;
  __syncthreads();

  const float4 ab = ((const float4*)anchors)[a];
  const float aarea = clip0(ab.z - ab.x) * clip0(ab.w - ab.y);
  float bi = -1.f;
  int bg = 0;
  const unsigned long long ainv = (unsigned long long)(0xFFFFFFFFu - (unsigned)a);
#pragma unroll 4
  for (int g = 0; g < GG; ++g) {
    float gx0 = sgt[g * 4 + 0], gy0 = sgt[g * 4 + 1];
    float gx1 = sgt[g * 4 + 2], gy1 = sgt[g * 4 + 3];
    float iw = clip0(fminf(ab.z, gx1) - fmaxf(ab.x, gx0));
    float ih = clip0(fminf(ab.w, gy1) - fmaxf(ab.y, gy0));
    float inter = iw * ih;
    float uni = aarea + clip0(gx1 - gx0) * clip0(gy1 - gy0) - inter;
    float iou = inter / (uni + EPSF);
    if (iou > bi) { bi = iou; bg = g; }  // strict > keeps first max (jnp.argmax)
    unsigned long long cand = ((unsigned long long)__float_as_uint(iou) << 32) | ainv;
    atomicMax(&slot[g], cand);
  }
  const size_t idx = (size_t)b * AA + a;
  best_iou[idx] = bi;
  best_idx[idx] = bg;
  __syncthreads();
  if (tid < GG) atomicMax(&best_anchor[b * GG + tid], slot[tid]);
}

// ---------------------------------------------------------------------------
// K2: focal loss on all anchors, DIoU on positives, negative-mining sort keys,
// and deterministic per-block partial sums (loc, pos_conf, n_pos).
// 256 threads x 16 anchors per block, 16 blocks per image.
// ---------------------------------------------------------------------------
__device__ __forceinline__ float focal_loss(float x, float t) {
  float p = 1.f / (1.f + expf(-x));
  float ce = fmaxf(x, 0.f) - x * t + log1pf(expf(-fabsf(x)));
  float pt = t * p + (1.f - t) * (1.f - p);
  float at = t * 0.25f + (1.f - t) * 0.75f;
  float om = 1.f - pt;
  return at * om * om * ce;  // gamma = 2
}

__global__ void k_loss(const float* __restrict__ bbox,
                       const float* __restrict__ conf,
                       const float* __restrict__ gt,
                       const float* __restrict__ best_iou,
                       const int* __restrict__ best_idx,
                       const unsigned long long* __restrict__ best_anchor,
                       unsigned* __restrict__ key,
                       float* __restrict__ partials) {
  const int b = blockIdx.y, blk = blockIdx.x, tid = threadIdx.x;
  __shared__ float sgt[GG * 4];
  __shared__ int winner[GG];
  __shared__ float r0[256], r1[256], r2[256];
  if (tid < GG * 4) sgt[tid] = gt[(size_t)b * GG * 4 + tid];
  if (tid < GG)
    winner[tid] =
        (int)(0xFFFFFFFFu - (unsigned)(best_anchor[b * GG + tid] & 0xFFFFFFFFull));
  __syncthreads();

  float loc_acc = 0.f, pos_acc = 0.f, np_acc = 0.f;
  for (int i = 0; i < 16; ++i) {
    const int a = blk * 4096 + i * 256 + tid;
    const size_t idx = (size_t)b * AA + a;
    __builtin_prefetch(conf + idx + 256, 0, 3);  // -> global_prefetch_b8
    float biou = best_iou[idx];
    bool pos = biou > 0.5f;
    for (int g = 0; g < GG && !pos; ++g) pos = (winner[g] == a);
    const float t = pos ? 1.f : 0.f;
    const float cl = focal_loss(conf[idx], t);
    if (pos) {
      np_acc += 1.f;
      pos_acc += fminf(cl, 100.f);
      const float4 p = ((const float4*)bbox)[idx];
      const int g = best_idx[idx];
      float gx0 = sgt[g * 4 + 0], gy0 = sgt[g * 4 + 1];
      float gx1 = sgt[g * 4 + 2], gy1 = sgt[g * 4 + 3];
      float iw = clip0(fminf(p.z, gx1) - fmaxf(p.x, gx0));
      float ih = clip0(fminf(p.w, gy1) - fmaxf(p.y, gy0));
      float inter = iw * ih;
      float uni = clip0(p.z - p.x) * clip0(p.w - p.y) +
                  clip0(gx1 - gx0) * clip0(gy1 - gy0) - inter;
      float iou = inter / (uni + EPSF);
      float ew = clip0(fmaxf(p.z, gx1) - fminf(p.x, gx0));
      float eh = clip0(fmaxf(p.w, gy1) - fminf(p.y, gy0));
      float c2 = ew * ew + eh * eh + EPSF;
      float dcx = 0.5f * ((p.x + p.z) - (gx0 + gx1));
      float dcy = 0.5f * ((p.y + p.w) - (gy0 + gy1));
      float d2 = dcx * dcx + dcy * dcy;
      loc_acc += fminf(1.f - iou + d2 / c2, 100.f);
      key[idx] = 0u;  // positives excluded from negative mining
    } else {
      // cl >= 0, so bits are order-preserving; +1 so real 0.0 beats positives
      key[idx] = __float_as_uint(cl) + 1u;
    }
  }
  r0[tid] = loc_acc; r1[tid] = pos_acc; r2[tid] = np_acc;
  __syncthreads();
  for (int s = 128; s > 0; s >>= 1) {
    if (tid < s) { r0[tid] += r0[tid + s]; r1[tid] += r1[tid + s]; r2[tid] += r2[tid + s]; }
    __syncthreads();
  }
  if (tid == 0) {
    float* p = partials + ((size_t)b * 16 + blk) * 3;
    p[0] = r0[0]; p[1] = r1[0]; p[2] = r2[0];
  }
}

// ---------------------------------------------------------------------------
// K3: WMMA reduction of partials. partials form exact 16(image)x16(block)
// matrices; D = A(16x4) * ones(4x16) + C chained over 4 chunks gives row sums
// (per-image totals) in every column of D. f32 WMMA -> full precision.
// A-matrix layout (ISA 7.12.2): lanes0-15 hold K={0,1}, lanes16-31 K={2,3}.
// D layout: VGPR r / lane 0 -> (M=r, N=0); VGPR r / lane 16 -> (M=r+8, N=0).
// ---------------------------------------------------------------------------
__global__ void k_reduce_wmma(const float* __restrict__ partials,
                              float* __restrict__ img_loc,
                              float* __restrict__ img_pos,
                              float* __restrict__ img_npos) {
  const int lane = threadIdx.x;        // exactly one wave32, EXEC all ones
  const int row = lane & 15;           // M = image
  const int khalf = (lane >> 4) * 2;   // K offset 0 or 2
  v2f ones; ones.x = 1.f; ones.y = 1.f;
  v8f acc0 = {}, acc1 = {}, acc2 = {};
#pragma unroll
  for (int ch = 0; ch < 4; ++ch) {
    const int k0 = ch * 4 + khalf;
    v2f a0, a1, a2;
    a0.x = partials[(row * 16 + k0) * 3 + 0]; a0.y = partials[(row * 16 + k0 + 1) * 3 + 0];
    a1.x = partials[(row * 16 + k0) * 3 + 1]; a1.y = partials[(row * 16 + k0 + 1) * 3 + 1];
    a2.x = partials[(row * 16 + k0) * 3 + 2]; a2.y = partials[(row * 16 + k0 + 1) * 3 + 2];
    acc0 = __builtin_amdgcn_wmma_f32_16x16x4_f32(false, a0, false, ones, (short)0, acc0, false, false);
    acc1 = __builtin_amdgcn_wmma_f32_16x16x4_f32(false, a1, false, ones, (short)0, acc1, false, false);
    acc2 = __builtin_amdgcn_wmma_f32_16x16x4_f32(false, a2, false, ones, (short)0, acc2, false, false);
  }
  if (lane == 0) {
#pragma unroll
    for (int r = 0; r < 8; ++r) { img_loc[r] = acc0[r]; img_pos[r] = acc1[r]; img_npos[r] = acc2[r]; }
  }
  if (lane == 16) {
#pragma unroll
    for (int r = 0; r < 8; ++r) { img_loc[8 + r] = acc0[r]; img_pos[8 + r] = acc1[r]; img_npos[8 + r] = acc2[r]; }
  }
}

// ---------------------------------------------------------------------------
// K4: exact top-k sum of negative focal losses per image (replaces the
// reference's full sort). 4-level 256-bin radix select on the uint keys,
// then one sum pass + tie correction at the threshold. Deterministic.
// ---------------------------------------------------------------------------
__global__ void k_negsel(const unsigned* __restrict__ key,
                         const float* __restrict__ img_npos,
                         float* __restrict__ img_neg) {
  const int b = blockIdx.x;
  const unsigned* k = key + (size_t)b * AA;
  const int tid = threadIdx.x;
  const int nt = blockDim.x;
  __shared__ unsigned hist[256];
  __shared__ unsigned s_prefix, s_remaining;
  __shared__ float red[1024];

  const int n_pos = (int)(img_npos[b] + 0.5f);
  int kt = AA - n_pos;
  if (n_pos * 3 < kt) kt = n_pos * 3;  // num_neg = min(A - n_pos, 3*n_pos)
  if (kt <= 0) { if (tid == 0) img_neg[b] = 0.f; return; }

  unsigned remaining = (unsigned)kt;
  unsigned prefix = 0;
  for (int level = 0; level < 4; ++level) {
    const int shift = 24 - level * 8;
    for (int i = tid; i < 256; i += nt) hist[i] = 0;
    __syncthreads();
    for (int i = tid; i < AA; i += nt) {
      unsigned kk = k[i];
      if (level == 0 || (kk >> (shift + 8)) == prefix)
        atomicAdd(&hist[(kk >> shift) & 255u], 1u);
    }
    __syncthreads();
    if (tid == 0) {
      unsigned run = 0;
      int d = 255;
      for (; d > 0; --d) {
        if (run + hist[d] >= remaining) break;
        run += hist[d];
      }
      s_prefix = (prefix << 8) | (unsigned)d;
      s_remaining = remaining - run;
    }
    __syncthreads();
    prefix = s_prefix;
    remaining = s_remaining;
    __syncthreads();
  }
  // prefix == exact threshold key T; take `remaining` copies of T.
  const unsigned T = prefix;
  const float tval = fminf(__uint_as_float(T - 1u), 100.f);
  float sum = 0.f;
  for (int i = tid; i < AA; i += nt) {
    unsigned kk = k[i];
    if (kk > T) sum += fminf(__uint_as_float(kk - 1u), 100.f);
  }
  red[tid] = sum;
  __syncthreads();
  for (int s = nt >> 1; s > 0; s >>= 1) {
    if (tid < s) red[tid] += red[tid + s];
    __syncthreads();
  }
  if (tid == 0) img_neg[b] = red[0] + (float)remaining * tval;
}

// ---------------------------------------------------------------------------
// K5: finalize the 3 scalars.
// ---------------------------------------------------------------------------
__global__ void k_final(const float* __restrict__ img_loc,
                        const float* __restrict__ img_pos,
                        const float* __restrict__ img_neg,
                        const float* __restrict__ img_npos,
                        float* __restrict__ out) {
  if (threadIdx.x == 0) {
    float l = 0.f, c = 0.f, np = 0.f;
    for (int b = 0; b < BB; ++b) {
      l += img_loc[b];
      c += img_pos[b] + img_neg[b];
      np += img_npos[b];
    }
    const float denom = fmaxf(1.f, np);
    const float loc = l / denom, conf = c / denom;
    out[0] = loc + conf;
    out[1] = conf;
    out[2] = loc;
  }
}

// ---------------------------------------------------------------------------
extern "C" void kernel_launch(void* const* d_in, const int* in_sizes, int n_in,
                              void* d_out, int out_size, void* d_ws, size_t ws_size,
                              hipStream_t stream) {
  const float* bbox = (const float*)d_in[0];     // [B,A,4]
  const float* conf = (const float*)d_in[1];     // [B,A]
  const float* anchors = (const float*)d_in[2];  // [A,4]
  const float* gt = (const float*)d_in[3];       // [B,G,4]

  char* ws = (char*)d_ws;
  unsigned long long* best_anchor = (unsigned long long*)ws;           // 4 KB
  float* best_iou = (float*)(ws + 4096);                               // 4 MB
  int* best_idx = (int*)(ws + 4096 + 4ull * BB * AA);                  // 4 MB
  unsigned* key = (unsigned*)(ws + 4096 + 8ull * BB * AA);             // 4 MB
  float* partials = (float*)(ws + 4096 + 12ull * BB * AA);             // 3 KB
  float* img_loc = partials + BB * 16 * 3;
  float* img_pos = img_loc + BB;
  float* img_npos = img_pos + BB;
  float* img_neg = img_npos + BB;

  k_init<<<1, 512, 0, stream>>>(best_anchor);
  k_match<<<dim3(AA / 256, BB), 256, 0, stream>>>(anchors, gt, best_iou, best_idx,
                                                  best_anchor);
  k_loss<<<dim3(16, BB), 256, 0, stream>>>(bbox, conf, gt, best_iou, best_idx,
                                           best_anchor, key, partials);
  k_reduce_wmma<<<1, 32, 0, stream>>>(partials, img_loc, img_pos, img_npos);
  k_negsel<<<BB, 1024, 0, stream>>>(key, img_npos, img_neg);
  k_final<<<1, 32, 0, stream>>>(img_loc, img_pos, img_neg, img_npos, (float*)d_out);
}